// G2P_6090263626287
// MI455X (gfx1250) — compile-verified
//
#include <hip/hip_runtime.h>
#include <hip/hip_bf16.h>

// ---------------------------------------------------------------------------
// G2P beam-search decoder for MI455X (gfx1250, wave32).
// fp32 WMMA (V_WMMA_F32_16X16X4_F32) for all decoder GEMMs; out_W (100MB)
// lives in the 192MB L2 across the 32 sequential decode steps.
// ---------------------------------------------------------------------------

typedef __attribute__((ext_vector_type(2))) float v2f;
typedef __attribute__((ext_vector_type(8))) float v8f;

#define HDIM   500
#define HPAD   512
#define G4H    2000
#define SEQ    32
#define BEAMN  10
#define MROWS  16
#define VOCAB  50000
#define NEGBIG (-1.0e30f)
#define FNEG   (-3.402823466e38f)

#if __has_builtin(__builtin_amdgcn_wmma_f32_16x16x4_f32)
#define HAVE_WMMA_F32 1
#else
#define HAVE_WMMA_F32 0
#endif

__device__ __forceinline__ float sigf(float x) { return 1.0f / (1.0f + __expf(-x)); }

// ---------------------------------------------------------------------------
// Generic skinny GEMM:  out(16 x N) (+)= A(16 x 512, LDS-staged, zero padded)
//                                       x B,  where B[k][n] = W[n*ldw + k]
// One 16-wide N tile per wave32. fp32 WMMA 16x16x4.
// REQUIREMENTS: Kreal % 4 == 0; W rows 8B-aligned (ldw even or *4 mult of 8);
// partial N tiles handled by clamping the B row (garbage lands in dead pad).
// ---------------------------------------------------------------------------
__global__ void __launch_bounds__(256)
wmma_gemm_k512(const float* __restrict__ A,
               const float* __restrict__ W,
               float* __restrict__ out,
               const float* __restrict__ bias,
               int N, int ldw, int Kreal, int ldout,
               int accumulate)
{
    __shared__ float Alds[MROWS * HPAD];
    for (int i = threadIdx.x; i < MROWS * HPAD; i += blockDim.x)
        Alds[i] = A[i];
    __syncthreads();

    const int lane  = threadIdx.x & 31;
    const int wave  = threadIdx.x >> 5;
    const int ntile = (N + 15) >> 4;
    const int tile  = blockIdx.x * (blockDim.x >> 5) + wave;
    if (tile >= ntile) return;

    const int col0 = tile << 4;
    const int m    = lane & 15;             // A row / D column within tile
    const int hi   = lane >> 4;             // 0: lanes 0-15, 1: lanes 16-31
    const int n    = col0 + m;              // global output column
    const int nw   = (n < N) ? n : (N - 1); // clamped for B reads (dead cols)

    const float* __restrict__ wrow = W + (size_t)nw * ldw;
    const float* __restrict__ arow = Alds + m * HPAD + (hi << 1);

    // warm the B row through L2 (unguarded, clamped addresses are valid)
    for (int pk = hi << 6; pk < Kreal; pk += 128)
        __builtin_prefetch(wrow + pk, 0, 1);

    v8f c = {};
    if (accumulate) {
#pragma unroll
        for (int r = 0; r < 8; ++r)
            c[r] = out[(r + (hi << 3)) * ldout + n];
    }

#if HAVE_WMMA_F32
    // Lane holds A[m][kk+2hi .. +1] and B[kk+2hi .. +1][col m]; zero-padded A
    // beyond Kreal contributes nothing, so the loop simply stops at Kreal.
#pragma unroll 5
    for (int kk = 0; kk < Kreal; kk += 4) {
        v2f a = *(const v2f*)(arow + kk);                 // ds_load_b64
        v2f b = *(const v2f*)(wrow + kk + (hi << 1));     // global_load_b64
        c = __builtin_amdgcn_wmma_f32_16x16x4_f32(
                /*neg_a=*/false, a, /*neg_b=*/false, b,
                /*c_mod=*/(short)0, c, /*reuse_a=*/false, /*reuse_b=*/false);
    }
#else
    for (int r = 0; r < 8; ++r) {
        int row = r + (hi << 3);
        float acc = c[r];
        for (int k = 0; k < Kreal; ++k)
            acc += Alds[row * HPAD + k] * wrow[k];
        c[r] = acc;
    }
#endif

#pragma unroll
    for (int r = 0; r < 8; ++r) {
        float v = c[r];
        if (bias) v += bias[n];   // bias only used when N is a multiple of 16
        out[(r + (hi << 3)) * ldout + n] = v;
    }
}

// ---------------------------------------------------------------------------
// Encoder: 32 sequential LSTM steps, M=1 mat-vec -> single 1024-thread block.
// ---------------------------------------------------------------------------
__global__ void enc_kernel(const int* __restrict__ g_seq,
                           const float* __restrict__ emb,
                           const float* __restrict__ Wih,
                           const float* __restrict__ Whh,
                           const float* __restrict__ bih,
                           const float* __restrict__ bhh,
                           float* __restrict__ ctx,
                           float* __restrict__ ench,
                           float* __restrict__ encc)
{
    __shared__ float x[HDIM];
    __shared__ float h[HDIM];
    __shared__ float cc[HDIM];
    __shared__ float g[G4H];
    const int tid = threadIdx.x;

    for (int i = tid; i < HDIM; i += blockDim.x) { h[i] = 0.0f; cc[i] = 0.0f; }
    __syncthreads();

    for (int t = 0; t < SEQ; ++t) {
        const int tok = g_seq[t];
        for (int i = tid; i < HDIM; i += blockDim.x) x[i] = emb[(size_t)tok * HDIM + i];
        __syncthreads();
        for (int nx = tid; nx < G4H; nx += blockDim.x) {
            const float* wi = Wih + (size_t)nx * HDIM;
            const float* wh = Whh + (size_t)nx * HDIM;
            float s = bih[nx] + bhh[nx];
            for (int k = 0; k < HDIM; ++k) s += wi[k] * x[k] + wh[k] * h[k];
            g[nx] = s;
        }
        __syncthreads();
        for (int i = tid; i < HDIM; i += blockDim.x) {
            float gi = g[i], gf = g[HDIM + i], gg = g[2 * HDIM + i], go = g[3 * HDIM + i];
            float c2 = sigf(gf) * cc[i] + sigf(gi) * tanhf(gg);
            float h2 = sigf(go) * tanhf(c2);
            cc[i] = c2;
            h[i]  = h2;
            ctx[t * HDIM + i] = h2;
        }
        __syncthreads();
    }
    for (int i = tid; i < HDIM; i += blockDim.x) { ench[i] = h[i]; encc[i] = cc[i]; }
}

// ---------------------------------------------------------------------------
// Init beam state: broadcast encoder (h,c) to 10 beams in padded 16x512 form.
// ---------------------------------------------------------------------------
__global__ void init_dec(const float* __restrict__ ench,
                         const float* __restrict__ encc,
                         float* __restrict__ h_pad, float* __restrict__ c_pad,
                         float* __restrict__ scores, int* __restrict__ tokens)
{
    const int gid = blockIdx.x * blockDim.x + threadIdx.x;
    if (gid < MROWS * HPAD) {
        const int b = gid >> 9, k = gid & (HPAD - 1);
        const bool live = (b < BEAMN) && (k < HDIM);
        h_pad[gid] = live ? ench[k] : 0.0f;
        c_pad[gid] = live ? encc[k] : 0.0f;
    }
    if (gid < MROWS) {
        scores[gid] = 0.0f;
        tokens[gid] = (gid == 0) ? 2 /*BOS*/ : 1 /*PAD*/;
    }
}

// ---------------------------------------------------------------------------
// Decoder-step helpers
// ---------------------------------------------------------------------------
__global__ void embed_dec(const float* __restrict__ demb,
                          const int* __restrict__ tokens,
                          float* __restrict__ e_pad)
{
    const int gid = blockIdx.x * blockDim.x + threadIdx.x;
    if (gid >= MROWS * HPAD) return;
    const int b = gid >> 9, k = gid & (HPAD - 1);
    e_pad[gid] = (b < BEAMN && k < HDIM) ? demb[(size_t)tokens[b] * HDIM + k] : 0.0f;
}

__global__ void dec_cell(const float* __restrict__ gates,   // 16 x 2000
                         const float* __restrict__ bih, const float* __restrict__ bhh,
                         const float* __restrict__ c_pad,
                         float* __restrict__ h2_pad, float* __restrict__ c2_pad)
{
    const int gid = blockIdx.x * blockDim.x + threadIdx.x;
    if (gid >= MROWS * HPAD) return;
    const int b = gid >> 9, k = gid & (HPAD - 1);
    float h2 = 0.0f, c2 = 0.0f;
    if (b < BEAMN && k < HDIM) {
        const float* gr = gates + (size_t)b * G4H;
        float gi = gr[k]            + bih[k]            + bhh[k];
        float gf = gr[HDIM + k]     + bih[HDIM + k]     + bhh[HDIM + k];
        float gg = gr[2 * HDIM + k] + bih[2 * HDIM + k] + bhh[2 * HDIM + k];
        float go = gr[3 * HDIM + k] + bih[3 * HDIM + k] + bhh[3 * HDIM + k];
        c2 = sigf(gf) * c_pad[gid] + sigf(gi) * tanhf(gg);
        h2 = sigf(go) * tanhf(c2);
    }
    h2_pad[gid] = h2;
    c2_pad[gid] = c2;
}

// attention scores + softmax + weighted context, single block
__global__ void attn_kernel(const float* __restrict__ ctx,     // 32 x 500
                            const float* __restrict__ h2_pad,  // 16 x 512
                            float* __restrict__ wctx_pad)      // 16 x 512
{
    __shared__ float sc[BEAMN * SEQ];
    const int tid = threadIdx.x;
    if (tid < BEAMN * SEQ) {
        const int b = tid / SEQ, j = tid % SEQ;
        float s = 0.0f;
        for (int k = 0; k < HDIM; ++k) s += ctx[j * HDIM + k] * h2_pad[b * HPAD + k];
        sc[b * SEQ + j] = s;
    }
    __syncthreads();
    if (tid < BEAMN) {
        float mx = FNEG;
        for (int j = 0; j < SEQ; ++j) mx = fmaxf(mx, sc[tid * SEQ + j]);
        float sum = 0.0f;
        for (int j = 0; j < SEQ; ++j) { float e = __expf(sc[tid * SEQ + j] - mx); sc[tid * SEQ + j] = e; sum += e; }
        float inv = 1.0f / sum;
        for (int j = 0; j < SEQ; ++j) sc[tid * SEQ + j] *= inv;
    }
    __syncthreads();
    for (int i = tid; i < MROWS * HPAD; i += blockDim.x) {
        const int b = i >> 9, k = i & (HPAD - 1);
        float s = 0.0f;
        if (b < BEAMN && k < HDIM)
            for (int j = 0; j < SEQ; ++j) s += sc[b * SEQ + j] * ctx[j * HDIM + k];
        wctx_pad[i] = s;
    }
}

__global__ void tanh_inplace(float* __restrict__ p, int n)
{
    const int gid = blockIdx.x * blockDim.x + threadIdx.x;
    if (gid < n) p[gid] = tanhf(p[gid]);
}

// per-beam log-sum-exp over the 50000 logits (block per beam)
__global__ void row_lse(const float* __restrict__ logits,
                        float* __restrict__ rowmax, float* __restrict__ lse)
{
    __shared__ float red[512];
    const int b = blockIdx.x, tid = threadIdx.x;
    const float* row = logits + (size_t)b * VOCAB;
    float mx = FNEG;
    for (int i = tid; i < VOCAB; i += 512) mx = fmaxf(mx, row[i]);
    red[tid] = mx; __syncthreads();
    for (int s = 256; s > 0; s >>= 1) { if (tid < s) red[tid] = fmaxf(red[tid], red[tid + s]); __syncthreads(); }
    const float m = red[0];
    __syncthreads();
    float sum = 0.0f;
    for (int i = tid; i < VOCAB; i += 512) sum += __expf(row[i] - m);
    red[tid] = sum; __syncthreads();
    for (int s = 256; s > 0; s >>= 1) { if (tid < s) red[tid] += red[tid + s]; __syncthreads(); }
    if (tid == 0) { rowmax[b] = m; lse[b] = logf(red[0]); }
}

// ---------------------------------------------------------------------------
// Exact two-stage top-10 over beam_lk (10*50000), jax tie-break (lowest idx).
// ---------------------------------------------------------------------------
__device__ __forceinline__ void topk_insert(float v, int e, float lv[10], int li[10])
{
    int pos = 10;
    for (int r = 0; r < 10; ++r)
        if (v > lv[r] || (v == lv[r] && (unsigned)e < (unsigned)li[r])) { pos = r; break; }
    if (pos == 10) return;
    for (int r = 9; r > pos; --r) { lv[r] = lv[r - 1]; li[r] = li[r - 1]; }
    lv[pos] = v; li[pos] = e;
}

__device__ __forceinline__ void serial_select10(float* sv, int* si, int n,
                                                float* outv, int* outi)
{
    for (int r = 0; r < 10; ++r) {
        float bv = FNEG; int bi = -1; int bs = -1;
        for (int j = 0; j < n; ++j) {
            if (bs < 0 || sv[j] > bv ||
                (sv[j] == bv && (unsigned)si[j] < (unsigned)bi)) {
                bv = sv[j]; bi = si[j]; bs = j;
            }
        }
        outv[r] = bv; outi[r] = bi;
        sv[bs] = FNEG; si[bs] = -1;
    }
}

__global__ void topk_stage1(const float* __restrict__ logits,
                            const float* __restrict__ rowmax,
                            const float* __restrict__ lse,
                            const float* __restrict__ scores, int t,
                            float* __restrict__ cval, int* __restrict__ cidx)
{
    const int CH = 1000;                       // 500 blocks x 1000 = 500000
    const int start = blockIdx.x * CH;
    float lv[10]; int li[10];
    for (int r = 0; r < 10; ++r) { lv[r] = FNEG; li[r] = -1; }
    for (int i = threadIdx.x; i < CH; i += blockDim.x) {
        const int e = start + i;
        const int b = e / VOCAB, p = e - b * VOCAB;
        const float lp = logits[(size_t)b * VOCAB + p] - rowmax[b] - lse[b];
        const float v  = (t == 0) ? ((b == 0) ? lp : NEGBIG) : (lp + scores[b]);
        topk_insert(v, e, lv, li);
    }
    __shared__ float sv[256 * 10];
    __shared__ int   si[256 * 10];
    for (int r = 0; r < 10; ++r) { sv[threadIdx.x * 10 + r] = lv[r]; si[threadIdx.x * 10 + r] = li[r]; }
    __syncthreads();
    if (threadIdx.x == 0)
        serial_select10(sv, si, 256 * 10, cval + blockIdx.x * 10, cidx + blockIdx.x * 10);
}

__global__ void topk_stage2(const float* __restrict__ cval, const int* __restrict__ cidx,
                            float* __restrict__ best, int* __restrict__ bidx)
{
    const int NC = 5000;
    float lv[10]; int li[10];
    for (int r = 0; r < 10; ++r) { lv[r] = FNEG; li[r] = -1; }
    for (int i = threadIdx.x; i < NC; i += blockDim.x)
        topk_insert(cval[i], cidx[i], lv, li);
    __shared__ float sv[256 * 10];
    __shared__ int   si[256 * 10];
    for (int r = 0; r < 10; ++r) { sv[threadIdx.x * 10 + r] = lv[r]; si[threadIdx.x * 10 + r] = li[r]; }
    __syncthreads();
    if (threadIdx.x == 0)
        serial_select10(sv, si, 256 * 10, best, bidx);
}

// commit beam step: split idx, record backpointers, gather (h,c) by prev_k
__global__ void update_kernel(const float* __restrict__ best, const int* __restrict__ bidx,
                              int t, float* __restrict__ scores, int* __restrict__ tokens,
                              int* __restrict__ prevKs, int* __restrict__ nextYs,
                              const float* __restrict__ h2_pad, const float* __restrict__ c2_pad,
                              float* __restrict__ h_pad, float* __restrict__ c_pad)
{
    __shared__ int pk[BEAMN];
    const int tid = threadIdx.x;
    if (tid < BEAMN) {
        const int idx = bidx[tid];
        const int p = idx / VOCAB;
        const int y = idx - p * VOCAB;
        pk[tid] = p;
        prevKs[t * BEAMN + tid] = p;
        nextYs[t * BEAMN + tid] = y;
        tokens[tid] = y;
        scores[tid] = best[tid];
    }
    __syncthreads();
    for (int i = tid; i < MROWS * HPAD; i += blockDim.x) {
        const int b = i >> 9;
        float hv = 0.0f, cv = 0.0f;
        if (b < BEAMN) {
            const int s = pk[b];
            hv = h2_pad[s * HPAD + (i & (HPAD - 1))];
            cv = c2_pad[s * HPAD + (i & (HPAD - 1))];
        }
        h_pad[i] = hv;
        c_pad[i] = cv;
    }
}

// backtrack: hyps[k][t] (output as float per harness convention)
__global__ void traceback(const int* __restrict__ prevKs, const int* __restrict__ nextYs,
                          float* __restrict__ out)
{
    const int k = threadIdx.x;
    if (k >= BEAMN) return;
    int cur = k;
    for (int t = SEQ - 1; t >= 0; --t) {
        out[k * SEQ + t] = (float)nextYs[t * BEAMN + cur];
        cur = prevKs[t * BEAMN + cur];
    }
}

// ---------------------------------------------------------------------------
// Host orchestration
// ---------------------------------------------------------------------------
extern "C" void kernel_launch(void* const* d_in, const int* in_sizes, int n_in,
                              void* d_out, int out_size, void* d_ws, size_t ws_size,
                              hipStream_t stream)
{
    const int*   g_seq   = (const int*)  d_in[0];
    const float* enc_emb = (const float*)d_in[1];
    const float* enc_Wih = (const float*)d_in[2];
    const float* enc_Whh = (const float*)d_in[3];
    const float* enc_bih = (const float*)d_in[4];
    const float* enc_bhh = (const float*)d_in[5];
    const float* dec_emb = (const float*)d_in[6];
    const float* dec_Wih = (const float*)d_in[7];
    const float* dec_Whh = (const float*)d_in[8];
    const float* dec_bih = (const float*)d_in[9];
    const float* dec_bhh = (const float*)d_in[10];
    const float* attn_W  = (const float*)d_in[11];
    const float* out_W   = (const float*)d_in[12];
    const float* out_b   = (const float*)d_in[13];

    float* ws = (float*)d_ws;
    size_t o = 0;
    float* ctx      = ws + o; o += SEQ * HDIM;       // encoder context 32x500
    float* ench     = ws + o; o += HPAD;
    float* encc     = ws + o; o += HPAD;
    float* e_pad    = ws + o; o += MROWS * HPAD;
    float* h_pad    = ws + o; o += MROWS * HPAD;
    float* c_pad    = ws + o; o += MROWS * HPAD;
    float* h2_pad   = ws + o; o += MROWS * HPAD;
    float* c2_pad   = ws + o; o += MROWS * HPAD;
    float* wctx_pad = ws + o; o += MROWS * HPAD;
    float* o_raw    = ws + o; o += MROWS * HPAD;     // tanh-combine output, ld=512
    float* gates    = ws + o; o += MROWS * G4H;      // 16 x 2000
    float* logits   = ws + o; o += (size_t)MROWS * VOCAB; // 16 x 50000
    float* rowmax   = ws + o; o += MROWS;
    float* lse      = ws + o; o += MROWS;
    float* scores   = ws + o; o += MROWS;
    float* best     = ws + o; o += MROWS;
    float* cval     = ws + o; o += 5000;
    int*   cidx     = (int*)(ws + o); o += 5000;
    int*   bidx     = (int*)(ws + o); o += MROWS;
    int*   tokens   = (int*)(ws + o); o += MROWS;
    int*   prevKs   = (int*)(ws + o); o += SEQ * BEAMN;
    int*   nextYs   = (int*)(ws + o); o += SEQ * BEAMN;

    // ---- encoder (sequential, single block) ----
    enc_kernel<<<1, 1024, 0, stream>>>(g_seq, enc_emb, enc_Wih, enc_Whh,
                                       enc_bih, enc_bhh, ctx, ench, encc);
    init_dec<<<(MROWS * HPAD + 255) / 256, 256, 0, stream>>>(ench, encc, h_pad, c_pad,
                                                             scores, tokens);

    // ---- 32 beam-search decode steps ----
    for (int t = 0; t < SEQ; ++t) {
        embed_dec<<<(MROWS * HPAD + 255) / 256, 256, 0, stream>>>(dec_emb, tokens, e_pad);

        // gates = e @ Wih^T ; gates += h @ Whh^T        (WMMA, N=2000, K=500)
        wmma_gemm_k512<<<16, 256, 0, stream>>>(e_pad, dec_Wih, gates, nullptr,
                                               G4H, HDIM, HDIM, G4H, 0);
        wmma_gemm_k512<<<16, 256, 0, stream>>>(h_pad, dec_Whh, gates, nullptr,
                                               G4H, HDIM, HDIM, G4H, 1);
        dec_cell<<<(MROWS * HPAD + 255) / 256, 256, 0, stream>>>(gates, dec_bih, dec_bhh,
                                                                 c_pad, h2_pad, c2_pad);

        attn_kernel<<<1, 1024, 0, stream>>>(ctx, h2_pad, wctx_pad);

        // o = tanh([h2 | wctx] @ attn_W^T): two K=500 WMMA passes over attn_W
        wmma_gemm_k512<<<4, 256, 0, stream>>>(h2_pad,   attn_W,        o_raw, nullptr,
                                              HDIM, 2 * HDIM, HDIM, HPAD, 0);
        wmma_gemm_k512<<<4, 256, 0, stream>>>(wctx_pad, attn_W + HDIM, o_raw, nullptr,
                                              HDIM, 2 * HDIM, HDIM, HPAD, 1);
        tanh_inplace<<<(MROWS * HPAD + 255) / 256, 256, 0, stream>>>(o_raw, MROWS * HPAD);

        // logits = o @ out_W^T + out_b    (WMMA, N=50000 — out_W lives in L2)
        wmma_gemm_k512<<<(3125 + 7) / 8, 256, 0, stream>>>(o_raw, out_W, logits, out_b,
                                                           VOCAB, HDIM, HDIM, VOCAB, 0);

        row_lse<<<BEAMN, 512, 0, stream>>>(logits, rowmax, lse);
        topk_stage1<<<500, 256, 0, stream>>>(logits, rowmax, lse, scores, t, cval, cidx);
        topk_stage2<<<1, 256, 0, stream>>>(cval, cidx, best, bidx);
        update_kernel<<<1, 512, 0, stream>>>(best, bidx, t, scores, tokens,
                                             prevKs, nextYs, h2_pad, c2_pad, h_pad, c_pad);
    }

    traceback<<<1, 32, 0, stream>>>(prevKs, nextYs, (float*)d_out);
}